// EdgeEmbeddingBlock_15307263443258
// MI455X (gfx1250) — compile-verified
//
#include <hip/hip_runtime.h>
#include <cstdint>

typedef __attribute__((ext_vector_type(2))) float v2f;
typedef __attribute__((ext_vector_type(8))) float v8f;

#define NEDGES   60000
#define KATTR    10
#define NMAXR    8
#define JDIM     16
#define PER_EDGE (KATTR * NMAXR * JDIM)   // 1280 floats per output per edge

__device__ __forceinline__ float sel16(int j,
    float a0, float a1, float a2, float a3, float a4, float a5, float a6, float a7,
    float a8, float a9, float a10, float a11, float a12, float a13, float a14, float a15)
{
    // binary-tree select: 15 v_cndmask, depth 4, no scratch
    float b0 = (j & 1) ? a1  : a0;
    float b1 = (j & 1) ? a3  : a2;
    float b2 = (j & 1) ? a5  : a4;
    float b3 = (j & 1) ? a7  : a6;
    float b4 = (j & 1) ? a9  : a8;
    float b5 = (j & 1) ? a11 : a10;
    float b6 = (j & 1) ? a13 : a12;
    float b7 = (j & 1) ? a15 : a14;
    float c0 = (j & 2) ? b1 : b0;
    float c1 = (j & 2) ? b3 : b2;
    float c2 = (j & 2) ? b5 : b4;
    float c3 = (j & 2) ? b7 : b6;
    float d0 = (j & 4) ? c1 : c0;
    float d1 = (j & 4) ? c3 : c2;
    return (j & 8) ? d1 : d0;
}

__global__ __launch_bounds__(256) void edge_embed_wmma_kernel(
    const int*   __restrict__ edge_index,   // (2, E)
    const float* __restrict__ edge_len,     // (E, 1)
    const float* __restrict__ edge_vec,     // (E, 3)
    const float* __restrict__ node_attrs,   // (N_NODES, 10)
    float*       __restrict__ out,          // combined_r | combined_i
    int nE)
{
    const int wave = blockIdx.x * (blockDim.x >> 5) + (threadIdx.x >> 5);
    if (wave >= nE) return;                 // wave-uniform guard; EXEC stays full
    const int  lane = threadIdx.x & 31;
    const int  l15  = lane & 15;
    const bool lo   = lane < 16;

    // ---- per-edge scalars (wave-uniform, computed redundantly per lane) ----
    const float x   = edge_len[wave];
    const int   snd = edge_index[wave];     // row 0 = sender
    const float vx  = edge_vec[3 * wave + 0];
    const float vy  = edge_vec[3 * wave + 1];
    const float vz  = edge_vec[3 * wave + 2];

    // radial prefactor: sqrt(2/R_CUT) * poly_cutoff(x) / x
    const float u  = x * 0.2f;                  // x / R_CUT
    const float u2 = u * u, u3 = u2 * u, u6 = u3 * u3;
    float env = 1.0f - 28.0f * u6 + 48.0f * u6 * u - 21.0f * u6 * u2;
    env = (u < 1.0f) ? env : 0.0f;
    const float rad_scale = 0.63245553203f * env / x;

    // zonal (spherical) harmonics, L <= 3
    const float nrm = sqrtf(vx * vx + vy * vy + vz * vz);
    const float rin = 1.0f / fmaxf(nrm, 1e-9f);
    const float uxn = vx * rin, uyn = vy * rin, uzn = vz * rin;
    const float ct  = uzn;
    float st2 = 1.0f - ct * ct;
    st2 = fminf(fmaxf(st2, 0.0f), 1.0f);
    const float st  = sqrtf(st2);
    const float phi = atan2f(uyn, uxn);
    const float c1 = cosf(phi), s1 = sinf(phi);
    const float c2 = c1 * c1 - s1 * s1, s2 = 2.0f * s1 * c1;
    const float c3 = c2 * c1 - s2 * s1, s3 = s2 * c1 + c2 * s1;

    const float P11 = -st;
    const float P22 = 3.0f * st2;
    const float P33 = -15.0f * st2 * st;
    const float P10 = ct;
    const float P21 = -3.0f * ct * st;
    const float P32 = 15.0f * ct * st2;
    const float P20 = 0.5f * (3.0f * ct * ct - 1.0f);
    const float P30 = (5.0f * ct * P20 - 2.0f * P10) * (1.0f / 3.0f);
    const float P31 = (5.0f * ct * P21 - 3.0f * P11) * 0.5f;

    const float N00 = 0.28209479177f, N10 = 0.48860251190f, N11 = 0.34549414947f;
    const float N20 = 0.63078313051f, N21 = 0.25751613468f, N22 = 0.12875806734f;
    const float N30 = 0.74635266518f, N31 = 0.21545345607f, N32 = 0.06813236510f, N33 = 0.02781492157f;

    const float t11 = N11 * P11, t21 = N21 * P21, t22 = N22 * P22;
    const float t31 = N31 * P31, t32 = N32 * P32, t33 = N33 * P33;

    // columns j = (l, m) in l-major, m = -l..l order
    const float yr0  =  N00,        yi0  = 0.0f;
    const float yr1  = -t11 * c1,   yi1  =  t11 * s1;   // (1,-1)
    const float yr2  =  N10 * P10,  yi2  = 0.0f;        // (1, 0)
    const float yr3  =  t11 * c1,   yi3  =  t11 * s1;   // (1, 1)
    const float yr4  =  t22 * c2,   yi4  = -t22 * s2;   // (2,-2)
    const float yr5  = -t21 * c1,   yi5  =  t21 * s1;   // (2,-1)
    const float yr6  =  N20 * P20,  yi6  = 0.0f;        // (2, 0)
    const float yr7  =  t21 * c1,   yi7  =  t21 * s1;   // (2, 1)
    const float yr8  =  t22 * c2,   yi8  =  t22 * s2;   // (2, 2)
    const float yr9  = -t33 * c3,   yi9  =  t33 * s3;   // (3,-3)
    const float yr10 =  t32 * c2,   yi10 = -t32 * s2;   // (3,-2)
    const float yr11 = -t31 * c1,   yi11 =  t31 * s1;   // (3,-1)
    const float yr12 =  N30 * P30,  yi12 = 0.0f;        // (3, 0)
    const float yr13 =  t31 * c1,   yi13 =  t31 * s1;   // (3, 1)
    const float yr14 =  t32 * c2,   yi14 =  t32 * s2;   // (3, 2)
    const float yr15 =  t33 * c3,   yi15 =  t33 * s3;   // (3, 3)

    // B operand: lane j (j<16) holds y[j] in K-row 0; lanes 16..31 (K-row 2) = 0
    const float brs = sel16(l15, yr0, yr1, yr2, yr3, yr4, yr5, yr6, yr7,
                                 yr8, yr9, yr10, yr11, yr12, yr13, yr14, yr15);
    const float bis = sel16(l15, yi0, yi1, yi2, yi3, yi4, yi5, yi6, yi7,
                                 yi8, yi9, yi10, yi11, yi12, yi13, yi14, yi15);
    v2f BR; BR.x = lo ? brs : 0.0f; BR.y = 0.0f;
    v2f BI; BI.x = lo ? bis : 0.0f; BI.y = 0.0f;

    const float* nap = node_attrs + (size_t)snd * KATTR;

    // per-lane output base: D layout is M = vgpr + 8*(lane>=16), N = lane&15
    const size_t obase = (size_t)wave * PER_EDGE + (lo ? 0u : 128u) + (size_t)l15;
    float* outR = out + obase;
    float* outI = out + (size_t)nE * PER_EDGE + obase;

#pragma unroll
    for (int t = 0; t < 5; ++t) {
        const int M = t * 16 + l15;          // global row in the 80-row (k,i) space
        const int k = M >> 3;
        const int i = M & 7;
        const float w  = (float)(i + 1) * 0.62831853072f;   // (i+1)*pi/R_CUT
        const float av = rad_scale * sinf(w * x) * nap[k];

        v2f A; A.x = lo ? av : 0.0f; A.y = 0.0f;   // K-col 0 only
        v8f z = {};

        v8f dr = __builtin_amdgcn_wmma_f32_16x16x4_f32(
            false, A, false, BR, (short)0, z, false, false);
        v8f di = __builtin_amdgcn_wmma_f32_16x16x4_f32(
            false, A, false, BI, (short)0, z, false, false);

#pragma unroll
        for (int v = 0; v < 8; ++v) {
            outR[t * 256 + v * 16] = dr[v];
            outI[t * 256 + v * 16] = di[v];
        }
    }
}

extern "C" void kernel_launch(void* const* d_in, const int* in_sizes, int n_in,
                              void* d_out, int out_size, void* d_ws, size_t ws_size,
                              hipStream_t stream) {
    const int*   edge_index = (const int*)  d_in[0];   // (2, E) int32
    const float* edge_len   = (const float*)d_in[1];   // (E, 1)
    const float* edge_vec   = (const float*)d_in[2];   // (E, 3)
    const float* node_attrs = (const float*)d_in[3];   // (N_NODES, 10)
    float* out = (float*)d_out;

    const int nE = in_sizes[1];                        // E from edge_lenghts
    const int wavesPerBlock = 8;                       // 256 threads, wave32
    const int blocks = (nE + wavesPerBlock - 1) / wavesPerBlock;

    edge_embed_wmma_kernel<<<blocks, 256, 0, stream>>>(
        edge_index, edge_len, edge_vec, node_attrs, out, nE);
}